// Tox21GCNConv_77025943487112
// MI455X (gfx1250) — compile-verified
//
#include <hip/hip_runtime.h>

typedef float v2f __attribute__((ext_vector_type(2)));
typedef float v8f __attribute__((ext_vector_type(8)));

#define NN 100000
#define DD 128

// ---------------------------------------------------------------- utilities
__global__ void zero_f4_kernel(float4* __restrict__ p, int n4) {
  int i = blockIdx.x * blockDim.x + threadIdx.x;
  if (i < n4) p[i] = make_float4(0.f, 0.f, 0.f, 0.f);
}

__global__ void deg_init_kernel(float* __restrict__ deg, int n) {
  int i = blockIdx.x * blockDim.x + threadIdx.x;
  if (i < n) deg[i] = 1.0f;  // self loop
}

__global__ void deg_count_kernel(const int* __restrict__ dst, float* __restrict__ deg, int e) {
  int i = blockIdx.x * blockDim.x + threadIdx.x;
  if (i < e) atomicAdd(&deg[dst[i]], 1.0f);
}

__global__ void dinv_kernel(const float* __restrict__ deg, float* __restrict__ dinv, int n) {
  int i = blockIdx.x * blockDim.x + threadIdx.x;
  if (i < n) dinv[i] = rsqrtf(deg[i]);
}

// ---------------------------------------------------------------- WMMA GEMM
// h[N,128] = x[N,128] @ W[128,128], f32 end to end via V_WMMA_F32_16X16X4_F32.
// One wave per 16x16 output tile; block = 8 waves = one 16-row slab across D=128.
__global__ void gemm_wmma_kernel(const float* __restrict__ x,
                                 const float* __restrict__ W,
                                 float* __restrict__ h) {
  const int wave = threadIdx.x >> 5;   // 0..7 -> N tile
  const int lane = threadIdx.x & 31;
  const int hi   = lane >> 4;          // which K half-pair this lane holds
  const int l15  = lane & 15;
  const int m0   = blockIdx.x * 16;
  const int n0   = wave * 16;

  // A frag (16x4 f32): lanes 0-15 hold M=l15, K={k,k+1}; lanes 16-31 K={k+2,k+3}
  const float* aptr = x + (size_t)(m0 + l15) * DD + 2 * hi;
  // B frag (4x16 f32): lanes 0-15 hold K={k,k+1} row, N=l15; lanes 16-31 K={k+2,k+3}
  const float* bptr = W + (size_t)(2 * hi) * DD + n0 + l15;

  v8f acc = {0.f, 0.f, 0.f, 0.f, 0.f, 0.f, 0.f, 0.f};

  #pragma unroll
  for (int k = 0; k < DD; k += 4) {
    v2f a; a.x = aptr[k];          a.y = aptr[k + 1];
    v2f b; b.x = bptr[k * DD];     b.y = bptr[k * DD + DD];
    acc = __builtin_amdgcn_wmma_f32_16x16x4_f32(
        /*neg_a=*/false, a, /*neg_b=*/false, b,
        /*c_mod=*/(short)0, acc, /*reuse_a=*/false, /*reuse_b=*/false);
  }

  // C/D layout: VGPR r -> M = r + 8*hi, N = l15
  float* hout = h + (size_t)(m0 + 8 * hi) * DD + n0 + l15;
  #pragma unroll
  for (int r = 0; r < 8; ++r) hout[(size_t)r * DD] = acc[r];
}

// ---------------------------------------------------------------- edge scatter
// One wave (32 lanes) per edge; each lane moves 4 consecutive floats.
__global__ void scatter_kernel(const int* __restrict__ src, const int* __restrict__ dst,
                               const float* __restrict__ h, const float* __restrict__ dinv,
                               float* __restrict__ out, int e) {
  int gid  = blockIdx.x * blockDim.x + threadIdx.x;
  int edge = gid >> 5;
  int lane = gid & 31;
  if (edge >= e) return;
  int s = src[edge];
  int t = dst[edge];
  float w = dinv[s] * dinv[t];
  const float4 v = *(const float4*)(h + (size_t)s * DD + lane * 4);
  float* o = out + (size_t)t * DD + lane * 4;
  atomicAdd(o + 0, v.x * w);
  atomicAdd(o + 1, v.y * w);
  atomicAdd(o + 2, v.z * w);
  atomicAdd(o + 3, v.w * w);
}

// ---------------------------------------------------------------- finalize
// out = relu(out + h * dinv^2 (self loop) + b)
__global__ void finalize_kernel(float* __restrict__ out, const float* __restrict__ h,
                                const float* __restrict__ dinv, const float* __restrict__ b,
                                int n) {
  int gid = blockIdx.x * blockDim.x + threadIdx.x;
  if (gid >= n) return;
  int node = gid >> 7;   // / 128
  int c    = gid & 127;
  float dv = dinv[node];
  float v  = out[gid] + h[gid] * dv * dv + b[c];
  out[gid] = v > 0.f ? v : 0.f;
}

// ---------------------------------------------------------------- launcher
extern "C" void kernel_launch(void* const* d_in, const int* in_sizes, int n_in,
                              void* d_out, int out_size, void* d_ws, size_t ws_size,
                              hipStream_t stream) {
  const float* x    = (const float*)d_in[0];
  const int*   eidx = (const int*)d_in[1];   // [2, E] flat: row0 = src, row1 = dst
  const float* W    = (const float*)d_in[2];
  const float* bias = (const float*)d_in[3];
  float* out = (float*)d_out;

  const int E = in_sizes[1] / 2;
  const int* src = eidx;
  const int* dst = eidx + E;

  float* ws   = (float*)d_ws;
  float* h    = ws;                         // N*128 floats (51.2 MB)
  float* dinv = ws + (size_t)NN * DD;       // N floats
  float* deg  = dinv + NN;                  // N floats

  const int T = 256;

  // 1) zero the accumulator (d_out) and init degrees
  int n4 = NN * DD / 4;
  zero_f4_kernel<<<(n4 + T - 1) / T, T, 0, stream>>>((float4*)out, n4);
  deg_init_kernel<<<(NN + T - 1) / T, T, 0, stream>>>(deg, NN);

  // 2) count incoming edges
  deg_count_kernel<<<(E + T - 1) / T, T, 0, stream>>>(dst, deg, E);

  // 3) dinv = deg^-1/2
  dinv_kernel<<<(NN + T - 1) / T, T, 0, stream>>>(deg, dinv, NN);

  // 4) h = x @ W  (WMMA f32)
  gemm_wmma_kernel<<<NN / 16, T, 0, stream>>>(x, W, h);

  // 5) gather-scale-scatter over edges (wave per edge)
  long long threads = (long long)E * 32;
  int blocks = (int)((threads + T - 1) / T);
  scatter_kernel<<<blocks, T, 0, stream>>>(src, dst, h, dinv, out, E);

  // 6) self-loop + bias + relu, in place
  int n = NN * DD;
  finalize_kernel<<<(n + T - 1) / T, T, 0, stream>>>(out, h, dinv, bias, n);
}